// DynamicPillarFeatureNet_60395830116555
// MI455X (gfx1250) — compile-verified
//
#include <hip/hip_runtime.h>

typedef __attribute__((ext_vector_type(2))) float v2f;
typedef __attribute__((ext_vector_type(8))) float v8f;

#define NPTS   400000
#define GXC    352
#define GYC    400
#define BATCH  2
#define CANVAS (GYC * GXC)               // 140800
#define NSEG   (BATCH * CANVAS)          // 281600
#define COUT   64
#define EPSV   1e-3f
#define VXC    0.2f
#define VYC    0.2f
#define XOFF   0.1f                      // VX/2 + pcr[0]
#define YOFF   (-39.9f)                  // VY/2 + pcr[1]

static_assert(NPTS % 16 == 0, "tile tail handling removed; N must be a multiple of 16");

// ---------------------------------------------------------------------------
// Kernel 1: zero the scatter workspace (cnt + vsum) and the output canvas.
// ---------------------------------------------------------------------------
__global__ void zero_kernel(float* __restrict__ ws_region, long long nws,
                            float* __restrict__ out_region, long long nout) {
    long long i = (long long)blockIdx.x * blockDim.x + threadIdx.x;
    long long stride = (long long)gridDim.x * blockDim.x;
    for (long long j = i; j < nws; j += stride)  ws_region[j]  = 0.0f;
    for (long long j = i; j < nout; j += stride) out_region[j] = 0.0f;
}

// ---------------------------------------------------------------------------
// Kernel 2: fold BN (eval) into the linear weights AND pre-swizzle the WMMA
// B fragments into per-lane order so the main kernel loads them as b64.
//   s = gamma * rsqrt(var + eps);  t = beta - mean * s
//   Fragment pair for (chunk c, tile t, lane l):
//     half = l>>4, n = t*16 + (l&15), k0 = c*4 + 2*half
//     Wfrag[(c*4+t)*32 + l] = { Ws[k0][n], Ws[k0+1][n] },  Ws[k][n]=W[k][n]*s[n]
//   (k >= 9 rows are the K-padding and contribute 0).
// ---------------------------------------------------------------------------
__global__ void prep_kernel(const float* __restrict__ W,
                            const float* __restrict__ gamma,
                            const float* __restrict__ beta,
                            const float* __restrict__ rmean,
                            const float* __restrict__ rvar,
                            float2* __restrict__ Wfrag,
                            float*  __restrict__ tbias) {
    __shared__ float s_sh[COUT];
    int tid = threadIdx.x;                         // 0..383
    if (tid < COUT) {
        float s = gamma[tid] * rsqrtf(rvar[tid] + EPSV);
        s_sh[tid] = s;
        tbias[tid] = beta[tid] - rmean[tid] * s;
    }
    __syncthreads();

    int c3   = tid / 128;                          // chunk 0..2
    int rem  = tid % 128;
    int tile = rem / 32;                           // 0..3
    int lane = rem % 32;
    int half = lane >> 4;
    int n    = tile * 16 + (lane & 15);
    int k0   = c3 * 4 + 2 * half;
    float x = (k0     < 9) ? W[k0 * COUT + n]       * s_sh[n] : 0.0f;
    float y = (k0 + 1 < 9) ? W[(k0 + 1) * COUT + n] * s_sh[n] : 0.0f;
    Wfrag[(c3 * 4 + tile) * 32 + lane] = make_float2(x, y);
}

// ---------------------------------------------------------------------------
// Kernel 3: per-voxel count and xyz-sum (for the cluster mean).
// ---------------------------------------------------------------------------
__global__ void scatter_mean_kernel(const float* __restrict__ feats,
                                    const int*   __restrict__ coors,
                                    float* __restrict__ cnt,
                                    float* __restrict__ vsum) {
    int p = blockIdx.x * blockDim.x + threadIdx.x;
    if (p >= NPTS) return;
    int4 c4 = reinterpret_cast<const int4*>(coors)[p];   // (b, z, y, x)
    int idx = c4.x * CANVAS + c4.z * GXC + c4.w;
    float4 f = reinterpret_cast<const float4*>(feats)[p];
    atomicAdd(&cnt[idx], 1.0f);
    atomicAdd(&vsum[idx * 3 + 0], f.x);
    atomicAdd(&vsum[idx * 3 + 1], f.y);
    atomicAdd(&vsum[idx * 3 + 2], f.z);
}

// ---------------------------------------------------------------------------
// Kernel 4: main PFN — augment (9ch), WMMA 16x16x4_f32 GEMM (K padded to 12),
// ReLU, atomic-max scatter (int atomicMax on non-negative float bits).
// Each wave32 handles exactly 16 points -> 16x64 output tile.
// No tail: NPTS % 16 == 0, so all lanes are always valid (uniform EXEC).
// ---------------------------------------------------------------------------
__global__ void __launch_bounds__(256)
pfn_wmma_kernel(const float*  __restrict__ feats,
                const int*    __restrict__ coors,
                const float*  __restrict__ cnt,
                const float*  __restrict__ vsum,
                const float2* __restrict__ Wfrag,
                const float*  __restrict__ tbias,
                float* __restrict__ out) {
    const int lane = threadIdx.x & 31;
    const int wave = threadIdx.x >> 5;
    const int row  = lane & 15;      // point row within the 16-row tile
    const int half = lane >> 4;      // 0: K={0,1} side, 1: K={2,3} side

    const long long p = ((long long)blockIdx.x * 8 + wave) * 16 + row;

    // Gather point data (lanes 16..31 duplicate lanes 0..15's point — cheap).
    float4 f  = reinterpret_cast<const float4*>(feats)[p];
    int4   c4 = reinterpret_cast<const int4*>(coors)[p];
    const int idx = c4.x * CANVAS + c4.z * GXC + c4.w;

    float n   = fmaxf(cnt[idx], 1.0f);
    float inv = 1.0f / n;
    float mx  = vsum[idx * 3 + 0] * inv;
    float my  = vsum[idx * 3 + 1] * inv;
    float mz  = vsum[idx * 3 + 2] * inv;

    // Augmented 9-channel feature.
    float a0v = f.x, a1v = f.y, a2v = f.z, a3v = f.w;
    float a4v = f.x - mx, a5v = f.y - my, a6v = f.z - mz;
    float a7v = f.x - ((float)c4.w * VXC + XOFF);
    float a8v = f.y - ((float)c4.z * VYC + YOFF);

    // A fragments (16x4 f32): lane holds A[row][2*half + {0,1}] per chunk.
    v2f a0, a1, a2;
    a0.x = half ? a2v : a0v;  a0.y = half ? a3v : a1v;   // K = 0..3
    a1.x = half ? a6v : a4v;  a1.y = half ? a7v : a5v;   // K = 4..7
    a2.x = half ? 0.f : a8v;  a2.y = 0.f;                // K = 8..11 (padded)

    // B fragments: pre-swizzled, one coalesced b64 load per (chunk, tile).
    v2f b[3][4];
#pragma unroll
    for (int c3 = 0; c3 < 3; ++c3)
#pragma unroll
        for (int tile = 0; tile < 4; ++tile) {
            float2 w = Wfrag[(c3 * 4 + tile) * 32 + lane];
            b[c3][tile].x = w.x;
            b[c3][tile].y = w.y;
        }

    // Bias-seeded accumulators (bias depends only on the column this lane owns).
    const int col = row;
    v8f acc[4];
#pragma unroll
    for (int tile = 0; tile < 4; ++tile) {
        float tb = tbias[tile * 16 + col];
#pragma unroll
        for (int r = 0; r < 8; ++r) acc[tile][r] = tb;
    }

    // GEMM: 3 chained K-chunks per tile (EXEC uniform, all-1s).
#pragma unroll
    for (int tile = 0; tile < 4; ++tile) {
        acc[tile] = __builtin_amdgcn_wmma_f32_16x16x4_f32(
            false, a0, false, b[0][tile], (short)0, acc[tile], false, false);
        acc[tile] = __builtin_amdgcn_wmma_f32_16x16x4_f32(
            false, a1, false, b[1][tile], (short)0, acc[tile], false, false);
        acc[tile] = __builtin_amdgcn_wmma_f32_16x16x4_f32(
            false, a2, false, b[2][tile], (short)0, acc[tile], false, false);
    }

    // Voxel index of the 8 rows this lane owns in D (m = half*8 + r).
    int pidx[8];
#pragma unroll
    for (int r = 0; r < 8; ++r) pidx[r] = __shfl(idx, half * 8 + r, 32);

    // ReLU + unconditional atomic max scatter (values >= 0 -> int-bit max exact).
#pragma unroll
    for (int tile = 0; tile < 4; ++tile) {
#pragma unroll
        for (int r = 0; r < 8; ++r) {
            float v = fmaxf(acc[tile][r], 0.0f);
            atomicMax(reinterpret_cast<int*>(
                          &out[(long long)pidx[r] * COUT + tile * 16 + col]),
                      __float_as_int(v));
        }
    }
}

// ---------------------------------------------------------------------------
extern "C" void kernel_launch(void* const* d_in, const int* in_sizes, int n_in,
                              void* d_out, int out_size, void* d_ws, size_t ws_size,
                              hipStream_t stream) {
    const float* features = (const float*)d_in[0];   // [N,4]
    const int*   coors    = (const int*)d_in[1];     // [N,4] (b,z,y,x)
    const float* W        = (const float*)d_in[2];   // [9,64]
    const float* gamma    = (const float*)d_in[3];   // [64]
    const float* beta     = (const float*)d_in[4];   // [64]
    const float* rmean    = (const float*)d_in[5];   // [64]
    const float* rvar     = (const float*)d_in[6];   // [64]
    float*       out      = (float*)d_out;           // [NSEG,64]

    // Workspace (floats): cnt[NSEG] | vsum[NSEG*3] | Wfrag[12*32*2] | t[64]
    float*  ws    = (float*)d_ws;
    float*  cnt   = ws;
    float*  vsum  = ws + NSEG;
    float2* Wfrag = (float2*)(ws + (long long)NSEG * 4);   // 8B-aligned (NSEG*16 bytes)
    float*  tbias = ws + (long long)NSEG * 4 + 12 * 32 * 2;

    const long long nws  = (long long)NSEG * 4;            // cnt + vsum only
    const long long nout = (long long)NSEG * COUT;

    zero_kernel<<<2048, 256, 0, stream>>>(cnt, nws, out, nout);
    prep_kernel<<<1, 384, 0, stream>>>(W, gamma, beta, rmean, rvar, Wfrag, tbias);
    scatter_mean_kernel<<<(NPTS + 255) / 256, 256, 0, stream>>>(features, coors, cnt, vsum);

    const int waves  = NPTS / 16;             // 25000
    const int blocks = (waves + 7) / 8;       // 3125 (8 waves / 256-thread block)
    pfn_wmma_kernel<<<blocks, 256, 0, stream>>>(features, coors, cnt, vsum, Wfrag, tbias, out);
}